// RNNModel_90374701842459
// MI455X (gfx1250) — compile-verified
//
#include <hip/hip_runtime.h>

typedef __attribute__((ext_vector_type(2))) float v2f;
typedef __attribute__((ext_vector_type(8))) float v8f;

#define BB 256
#define SS 2048
#define II 32
#define HH 18

#define CH     32            // timesteps per async-streamed chunk
#define NCHUNK (SS / CH)     // 64
#define CHF    (CH * HH)     // 576 floats (2304 B) per chunk
#define NOPS   (CHF / 64)    // 9 async b64 ops per chunk (32 lanes x 8 B)

// ---------------------------------------------------------------------------
// Kernel 1: Y[M,18] = X[M,K] @ W^T + (bih + bhh), via V_WMMA_F32_16X16X4_F32
// f32 A 16x4 layout: lane = M row (lane&15), K = 4j + 2*(lane>>4) + reg.
// f32 D 16x16 layout: vgpr r = row r (lanes 0-15) / r+8 (lanes 16-31), lane = col.
// ---------------------------------------------------------------------------
template <int K>
__global__ __launch_bounds__(256) void gemm_in_kernel(
    const float* __restrict__ X, const float* __restrict__ W,
    const float* __restrict__ bih, const float* __restrict__ bhh,
    float* __restrict__ Y, int Mtiles)
{
    constexpr int KSTEPS = (K + 3) / 4;
    const int lane  = threadIdx.x & 31;
    const int l16   = lane & 15;
    const int half  = lane >> 4;
    const int wave  = (blockIdx.x * blockDim.x + threadIdx.x) >> 5;
    const int nwave = (gridDim.x * blockDim.x) >> 5;

    // Precompute B fragments (weights) once per wave; zero-pad K and N>=18.
    v2f bf[KSTEPS][2];
#pragma unroll
    for (int j = 0; j < KSTEPS; ++j) {
#pragma unroll
        for (int nt = 0; nt < 2; ++nt) {
            const int col = nt * 16 + l16;
            const int k0  = 4 * j + half * 2;
            float e0 = (col < HH && k0     < K) ? W[col * K + k0    ] : 0.f;
            float e1 = (col < HH && k0 + 1 < K) ? W[col * K + k0 + 1] : 0.f;
            v2f t; t.x = e0; t.y = e1;
            bf[j][nt] = t;
        }
    }
    const float bias0 = bih[l16] + bhh[l16];                 // cols 0..15 < 18
    const float bias1 = (16 + l16 < HH) ? (bih[16 + l16] + bhh[16 + l16]) : 0.f;

    for (int tile = wave; tile < Mtiles; tile += nwave) {
        const float* Xr = X + (size_t)(tile * 16 + l16) * K;
        v8f acc0 = {}; v8f acc1 = {};
#pragma unroll
        for (int j = 0; j < KSTEPS; ++j) {
            const int k0 = 4 * j + half * 2;
            v2f a;
            a.x = (k0     < K) ? Xr[k0    ] : 0.f;
            a.y = (k0 + 1 < K) ? Xr[k0 + 1] : 0.f;
            acc0 = __builtin_amdgcn_wmma_f32_16x16x4_f32(
                false, a, false, bf[j][0], (short)0, acc0, false, false);
            acc1 = __builtin_amdgcn_wmma_f32_16x16x4_f32(
                false, a, false, bf[j][1], (short)0, acc1, false, false);
        }
        const size_t rowbase = (size_t)tile * 16 + half * 8;
#pragma unroll
        for (int r = 0; r < 8; ++r)
            Y[(rowbase + r) * HH + l16] = acc0[r] + bias0;
        if (l16 < HH - 16) {
#pragma unroll
            for (int r = 0; r < 8; ++r)
                Y[(rowbase + r) * HH + 16 + l16] = acc1[r] + bias1;
        }
    }
}

// ---------------------------------------------------------------------------
// Kernel 2: sequential scan  h = tanh(xw_t + h @ Whh^T)
// One wave32 per batch element; lane j (<18) owns h_j. xW is streamed into
// LDS with double-buffered GLOBAL_LOAD_ASYNC_TO_LDS_B64 (ASYNCcnt), so the
// per-step critical path is only: v_tanh -> v_readlane -> 4 FMA chains.
// ---------------------------------------------------------------------------
__global__ __launch_bounds__(32) void rnn_scan_kernel(
    const float* __restrict__ xw,   // [B,S,H] pre-activations
    const float* __restrict__ Whh,  // [H,H]
    const float* __restrict__ h0,   // [B,H]
    float* __restrict__ y,          // [B,S,H] layer output
    float* __restrict__ hT)         // [B,H] final hidden
{
    __shared__ float sbuf[2][CHF];  // 2 x 2304 B double buffer
    const int b = blockIdx.x;
    const int j = threadIdx.x;      // lane
    const bool active = (j < HH);

    float w[HH];
#pragma unroll
    for (int k = 0; k < HH; ++k) w[k] = active ? Whh[j * HH + k] : 0.f;
    float h = active ? h0[b * HH + j] : 0.f;

    const float* xwb = xw + (size_t)b * SS * HH;
    float*       yb  = y  + (size_t)b * SS * HH;

    // Issue async copy of chunk c into sbuf[c & 1] (full wave, uniform flow).
    auto issue_chunk = [&](int c) {
        const float* src = xwb + (size_t)c * CHF;        // 8-B aligned
#pragma unroll
        for (int i = 0; i < NOPS; ++i) {
            unsigned     lds = (unsigned)(size_t)&sbuf[c & 1][i * 64 + j * 2];
            const float* g   = src + i * 64 + j * 2;
            asm volatile("global_load_async_to_lds_b64 %0, %1, off"
                         :: "v"(lds), "v"(g) : "memory");
        }
    };

    issue_chunk(0);
    for (int c = 0; c < NCHUNK; ++c) {
        if (c + 1 < NCHUNK) {
            issue_chunk(c + 1);                          // keep HBM busy
            asm volatile("s_wait_asynccnt 0x9" ::: "memory");  // chunk c done
        } else {
            asm volatile("s_wait_asynccnt 0x0" ::: "memory");
        }
        const float* sb = sbuf[c & 1];
        float xv = active ? sb[j] : 0.f;                 // preload t=0 of chunk
#pragma unroll 4
        for (int tt = 0; tt < CH; ++tt) {
            float nxt = 0.f;                             // preload next step
            if (tt + 1 < CH) nxt = active ? sb[(tt + 1) * HH + j] : 0.f;
            float acc[4] = {0.f, 0.f, 0.f, 0.f};
#pragma unroll
            for (int k = 0; k < HH; ++k) {
                float hk = __int_as_float(
                    __builtin_amdgcn_readlane(__float_as_int(h), k));
                acc[k & 3] = fmaf(w[k], hk, acc[k & 3]);
            }
            const float v = ((acc[0] + acc[1]) + (acc[2] + acc[3])) + xv;
            float th;
            asm volatile("v_tanh_f32 %0, %1\n\tv_nop" : "=v"(th) : "v"(v));
            h = th;
            if (active) yb[(size_t)(c * CH + tt) * HH + j] = h;
            xv = nxt;
        }
    }
    if (active) hT[b * HH + j] = h;
}

// ---------------------------------------------------------------------------
// Kernel 3: MLP head 18->18->10->8->4->2->1, weights staged in LDS.
// ---------------------------------------------------------------------------
__global__ __launch_bounds__(256) void mlp_kernel(
    const float* __restrict__ X,  // [M,18]
    const float* __restrict__ W1, const float* __restrict__ b1,
    const float* __restrict__ W2, const float* __restrict__ b2,
    const float* __restrict__ W3, const float* __restrict__ b3,
    const float* __restrict__ W4, const float* __restrict__ b4,
    const float* __restrict__ W5, const float* __restrict__ b5,
    const float* __restrict__ W6, const float* __restrict__ b6,
    float* __restrict__ out, int M)
{
    __shared__ float sW1[18*18], sb1[18];
    __shared__ float sW2[10*18], sb2[10];
    __shared__ float sW3[8*10],  sb3[8];
    __shared__ float sW4[4*8],   sb4[4];
    __shared__ float sW5[2*4],   sb5[2];
    __shared__ float sW6[2],     sb6[1];
    for (int i = threadIdx.x; i < 18*18; i += blockDim.x) sW1[i] = W1[i];
    for (int i = threadIdx.x; i < 10*18; i += blockDim.x) sW2[i] = W2[i];
    for (int i = threadIdx.x; i < 8*10;  i += blockDim.x) sW3[i] = W3[i];
    for (int i = threadIdx.x; i < 4*8;   i += blockDim.x) sW4[i] = W4[i];
    for (int i = threadIdx.x; i < 2*4;   i += blockDim.x) sW5[i] = W5[i];
    if (threadIdx.x < 18) sb1[threadIdx.x] = b1[threadIdx.x];
    if (threadIdx.x < 10) sb2[threadIdx.x] = b2[threadIdx.x];
    if (threadIdx.x < 8)  sb3[threadIdx.x] = b3[threadIdx.x];
    if (threadIdx.x < 4)  sb4[threadIdx.x] = b4[threadIdx.x];
    if (threadIdx.x < 2)  { sb5[threadIdx.x] = b5[threadIdx.x]; sW6[threadIdx.x] = W6[threadIdx.x]; }
    if (threadIdx.x < 1)  sb6[0] = b6[0];
    __syncthreads();

    for (int i = blockIdx.x * blockDim.x + threadIdx.x; i < M;
         i += gridDim.x * blockDim.x) {
        float a[18];
#pragma unroll
        for (int k = 0; k < 18; ++k) a[k] = X[(size_t)i * 18 + k];
        float t1[18];
#pragma unroll
        for (int o = 0; o < 18; ++o) {
            float s = sb1[o];
#pragma unroll
            for (int k = 0; k < 18; ++k) s = fmaf(sW1[o*18+k], a[k], s);
            t1[o] = fmaxf(s, 0.f);
        }
        float t2[10];
#pragma unroll
        for (int o = 0; o < 10; ++o) {
            float s = sb2[o];
#pragma unroll
            for (int k = 0; k < 18; ++k) s = fmaf(sW2[o*18+k], t1[k], s);
            t2[o] = fmaxf(s, 0.f);
        }
        float t3[8];
#pragma unroll
        for (int o = 0; o < 8; ++o) {
            float s = sb3[o];
#pragma unroll
            for (int k = 0; k < 10; ++k) s = fmaf(sW3[o*10+k], t2[k], s);
            t3[o] = fmaxf(s, 0.f);
        }
        float t4[4];
#pragma unroll
        for (int o = 0; o < 4; ++o) {
            float s = sb4[o];
#pragma unroll
            for (int k = 0; k < 8; ++k) s = fmaf(sW4[o*8+k], t3[k], s);
            t4[o] = fmaxf(s, 0.f);
        }
        float t5[2];
#pragma unroll
        for (int o = 0; o < 2; ++o) {
            float s = sb5[o];
#pragma unroll
            for (int k = 0; k < 4; ++k) s = fmaf(sW5[o*4+k], t4[k], s);
            t5[o] = fmaxf(s, 0.f);
        }
        out[i] = fmaf(sW6[0], t5[0], fmaf(sW6[1], t5[1], sb6[0]));
    }
}

// ---------------------------------------------------------------------------
extern "C" void kernel_launch(void* const* d_in, const int* in_sizes, int n_in,
                              void* d_out, int out_size, void* d_ws, size_t ws_size,
                              hipStream_t stream) {
    (void)in_sizes; (void)n_in; (void)out_size; (void)ws_size;
    const float* input  = (const float*)d_in[0];   // [B,S,I]
    const float* hidden = (const float*)d_in[1];   // [3,B,H]
    const float* Wih[3] = {(const float*)d_in[2],  (const float*)d_in[6],  (const float*)d_in[10]};
    const float* bih[3] = {(const float*)d_in[3],  (const float*)d_in[7],  (const float*)d_in[11]};
    const float* Whh[3] = {(const float*)d_in[4],  (const float*)d_in[8],  (const float*)d_in[12]};
    const float* bhh[3] = {(const float*)d_in[5],  (const float*)d_in[9],  (const float*)d_in[13]};
    const float* W1 = (const float*)d_in[14]; const float* b1 = (const float*)d_in[15];
    const float* W2 = (const float*)d_in[16]; const float* b2 = (const float*)d_in[17];
    const float* W3 = (const float*)d_in[18]; const float* b3 = (const float*)d_in[19];
    const float* W4 = (const float*)d_in[20]; const float* b4 = (const float*)d_in[21];
    const float* W5 = (const float*)d_in[22]; const float* b5 = (const float*)d_in[23];
    const float* W6 = (const float*)d_in[24]; const float* b6 = (const float*)d_in[25];

    float* out     = (float*)d_out;
    float* hid_out = out + (size_t)BB * SS;        // [3,B,H] tail of d_out

    const int    M      = BB * SS;                 // 524288
    const int    Mtiles = M / 16;                  // 32768
    float* buf0 = (float*)d_ws;                    // [M,H]
    float* buf1 = buf0 + (size_t)M * HH;           // [M,H]

    // Layer 0
    gemm_in_kernel<II><<<512, 256, 0, stream>>>(input, Wih[0], bih[0], bhh[0], buf0, Mtiles);
    rnn_scan_kernel<<<BB, 32, 0, stream>>>(buf0, Whh[0], hidden + 0 * BB * HH,
                                           buf1, hid_out + 0 * BB * HH);
    // Layer 1
    gemm_in_kernel<HH><<<512, 256, 0, stream>>>(buf1, Wih[1], bih[1], bhh[1], buf0, Mtiles);
    rnn_scan_kernel<<<BB, 32, 0, stream>>>(buf0, Whh[1], hidden + 1 * BB * HH,
                                           buf1, hid_out + 1 * BB * HH);
    // Layer 2
    gemm_in_kernel<HH><<<512, 256, 0, stream>>>(buf1, Wih[2], bih[2], bhh[2], buf0, Mtiles);
    rnn_scan_kernel<<<BB, 32, 0, stream>>>(buf0, Whh[2], hidden + 2 * BB * HH,
                                           buf1, hid_out + 2 * BB * HH);
    // MLP head
    mlp_kernel<<<1024, 256, 0, stream>>>(buf1, W1, b1, W2, b2, W3, b3,
                                         W4, b4, W5, b5, W6, b6, out, M);
}